// GraphSAGEModel_47115791237141
// MI455X (gfx1250) — compile-verified
//
#include <hip/hip_runtime.h>
#include <math.h>

typedef __attribute__((ext_vector_type(2))) float v2f;
typedef __attribute__((ext_vector_type(8))) float v8f;

static constexpr int Nn = 100000;
static constexpr int Ee = 1600000;

// Force the hardware fp32 atomic (global_atomic_add_f32, no CAS loop):
// relaxed + agent scope lowers to atomicrmw fadd syncscope("agent") monotonic.
__device__ __forceinline__ void atom_add_f32(float* p, float v) {
  __hip_atomic_fetch_add(p, v, __ATOMIC_RELAXED, __HIP_MEMORY_SCOPE_AGENT);
}

__global__ __launch_bounds__(256) void k_zero(float* __restrict__ p, int count) {
  int stride = gridDim.x * blockDim.x;
  for (int i = blockIdx.x * blockDim.x + threadIdx.x; i < count; i += stride)
    p[i] = 0.0f;
}

__global__ __launch_bounds__(256) void k_deg(const int* __restrict__ dst,
                                             float* __restrict__ deg, int E) {
  int e = blockIdx.x * 256 + threadIdx.x;
  if (e < E) atom_add_f32(&deg[dst[e]], 1.0f);
}

__global__ __launch_bounds__(256) void k_invdeg(float* __restrict__ deg, int n) {
  int i = blockIdx.x * 256 + threadIdx.x;
  if (i < n) deg[i] = 1.0f / fmaxf(deg[i], 1.0f);
}

// One thread per (edge, 4-channel group): float4 gather + 4 fp32 L2 atomics.
// C = 4 << GSHIFT channels per row.
template <int GSHIFT>
__global__ __launch_bounds__(256) void k_scatter(const float* __restrict__ feat,
                                                 const int* __restrict__ src,
                                                 const int* __restrict__ dst,
                                                 float* __restrict__ agg, int E) {
  const int GROUPS = 1 << GSHIFT;
  const int C = GROUPS * 4;
  unsigned tid = blockIdx.x * 256u + threadIdx.x;
  unsigned e = tid >> GSHIFT;
  if (e >= (unsigned)E) return;
  int c = (int)(tid & (GROUPS - 1)) * 4;
  int s = src[e], d = dst[e];
  const float4 v = *(const float4*)(feat + (size_t)s * C + c);
  float* o = agg + (size_t)d * C + c;
  atom_add_f32(o + 0, v.x);
  atom_add_f32(o + 1, v.y);
  atom_add_f32(o + 2, v.z);
  atom_add_f32(o + 3, v.w);
}

// Layer 1: h1 = relu([agg1*inv | x] @ [W1_l | W1_r]^T + b1)
// One wave computes one 16x16 tile of h1 via V_WMMA_F32_16X16X4_F32.
// A 16x4 f32 layout: lane = (k/2)*16 + m, vgpr = k%2.
// B 4x16 f32 layout: lane = (k/2)*16 + n, vgpr = k%2.
// C/D 16x16 f32 layout: vgpr r, lanes 0-15 -> M=r, lanes 16-31 -> M=r+8, N=lane%16.
__global__ __launch_bounds__(256) void k_gemm1(
    const float* __restrict__ agg1, const float* __restrict__ inv,
    const float* __restrict__ x, const float* __restrict__ W1l,
    const float* __restrict__ W1r, const float* __restrict__ b1,
    float* __restrict__ h1) {
  const int lane = threadIdx.x & 31;
  const int wave = threadIdx.x >> 5;  // 8 waves -> 8 col tiles (HID_C=128)
  const int row0 = blockIdx.x * 16;
  const int col0 = wave * 16;
  const int g = lane >> 4;            // selects K-pair within the K=4 step
  const int mn = lane & 15;
  const int m = row0 + mn;            // A row held by this lane
  const int n = col0 + mn;            // B col held by this lane
  const float invm = inv[m];
  const float* __restrict__ arow = agg1 + (size_t)m * 64;
  const float* __restrict__ xrow = x + (size_t)m * 64;
  const float* __restrict__ wl = W1l + (size_t)n * 64;
  const float* __restrict__ wr = W1r + (size_t)n * 64;
  v8f acc = {};
#pragma unroll
  for (int kk = 0; kk < 32; ++kk) {   // K = 64 (agg) + 64 (self), 4 per step
    const int k0 = kk * 4 + g * 2;
    v2f a, b;
    if (k0 < 64) {
      a.x = arow[k0] * invm;
      a.y = arow[k0 + 1] * invm;
      b.x = wl[k0];
      b.y = wl[k0 + 1];
    } else {
      const int k = k0 - 64;
      a.x = xrow[k];
      a.y = xrow[k + 1];
      b.x = wr[k];
      b.y = wr[k + 1];
    }
    acc = __builtin_amdgcn_wmma_f32_16x16x4_f32(false, a, false, b, (short)0,
                                                acc, false, false);
  }
  const float bias = b1[n];
#pragma unroll
  for (int r = 0; r < 8; ++r) {
    const int mm = row0 + r + g * 8;
    float v = acc[r] + bias;
    h1[(size_t)mm * 128 + col0 + mn] = fmaxf(v, 0.0f);
  }
}

// Layer 2: out = sigmoid([agg2*inv | h1] @ [W2_l | W2_r]^T + b2)
__global__ __launch_bounds__(128) void k_gemm2(
    const float* __restrict__ agg2, const float* __restrict__ inv,
    const float* __restrict__ h1, const float* __restrict__ W2l,
    const float* __restrict__ W2r, const float* __restrict__ b2,
    float* __restrict__ out) {
  const int lane = threadIdx.x & 31;
  const int wave = threadIdx.x >> 5;  // 4 waves -> 4 col tiles (OUT_C=64)
  const int row0 = blockIdx.x * 16;
  const int col0 = wave * 16;
  const int g = lane >> 4;
  const int mn = lane & 15;
  const int m = row0 + mn;
  const int n = col0 + mn;
  const float invm = inv[m];
  const float* __restrict__ arow = agg2 + (size_t)m * 128;
  const float* __restrict__ hrow = h1 + (size_t)m * 128;
  const float* __restrict__ wl = W2l + (size_t)n * 128;
  const float* __restrict__ wr = W2r + (size_t)n * 128;
  v8f acc = {};
#pragma unroll
  for (int kk = 0; kk < 64; ++kk) {   // K = 128 (agg) + 128 (self)
    const int k0 = kk * 4 + g * 2;
    v2f a, b;
    if (k0 < 128) {
      a.x = arow[k0] * invm;
      a.y = arow[k0 + 1] * invm;
      b.x = wl[k0];
      b.y = wl[k0 + 1];
    } else {
      const int k = k0 - 128;
      a.x = hrow[k];
      a.y = hrow[k + 1];
      b.x = wr[k];
      b.y = wr[k + 1];
    }
    acc = __builtin_amdgcn_wmma_f32_16x16x4_f32(false, a, false, b, (short)0,
                                                acc, false, false);
  }
  const float bias = b2[n];
#pragma unroll
  for (int r = 0; r < 8; ++r) {
    const int mm = row0 + r + g * 8;
    float v = acc[r] + bias;
    out[(size_t)mm * 64 + col0 + mn] = 1.0f / (1.0f + __expf(-v));
  }
}

extern "C" void kernel_launch(void* const* d_in, const int* in_sizes, int n_in,
                              void* d_out, int out_size, void* d_ws,
                              size_t ws_size, hipStream_t stream) {
  (void)in_sizes; (void)n_in; (void)out_size; (void)ws_size;
  const float* x   = (const float*)d_in[0];
  const int*   ei  = (const int*)d_in[1];   // [2,E] int32
  const float* W1l = (const float*)d_in[2];
  const float* W1r = (const float*)d_in[3];
  const float* b1  = (const float*)d_in[4];
  const float* W2l = (const float*)d_in[5];
  const float* W2r = (const float*)d_in[6];
  const float* b2  = (const float*)d_in[7];
  const int* src = ei;
  const int* dst = ei + Ee;

  // Workspace layout (floats): deg[N] | agg1[N*64] | agg2[N*128] | h1[N*128]
  float* ws   = (float*)d_ws;
  float* deg  = ws;
  float* agg1 = deg + Nn;
  float* agg2 = agg1 + (size_t)Nn * 64;
  float* h1   = agg2 + (size_t)Nn * 128;

  const int zeroCount = Nn * (1 + 64 + 128);  // deg|agg1|agg2 are contiguous
  k_zero<<<4096, 256, 0, stream>>>(deg, zeroCount);
  k_deg<<<(Ee + 255) / 256, 256, 0, stream>>>(dst, deg, Ee);
  k_scatter<4><<<(Ee * 16) / 256, 256, 0, stream>>>(x, src, dst, agg1, Ee);
  k_invdeg<<<(Nn + 255) / 256, 256, 0, stream>>>(deg, Nn);
  k_gemm1<<<Nn / 16, 256, 0, stream>>>(agg1, deg, x, W1l, W1r, b1, h1);
  k_scatter<5><<<(Ee * 32) / 256, 256, 0, stream>>>(h1, src, dst, agg2, Ee);
  k_gemm2<<<Nn / 16, 128, 0, stream>>>(agg2, deg, h1, W2l, W2r, b2,
                                       (float*)d_out);
}